// LossRegressionGaussianWithCorrelations_11312943857918
// MI455X (gfx1250) — compile-verified
//
#include <hip/hip_runtime.h>
#include <math.h>

typedef float v2f __attribute__((ext_vector_type(2)));
typedef float v4f __attribute__((ext_vector_type(4)));
typedef float v8f __attribute__((ext_vector_type(8)));

#define NB 16
#define MAXW 1024  // problem size: n_weights == 1024 (LDS slab sized for this)
#define LOG2PI 1.8378770664093454835606594728112353

__device__ inline v8f wmma_f32_k4(v2f a, v2f b, v8f c) {
  // D = A(16x4 f32) * B(4x16 f32) + C(16x16 f32)
  return __builtin_amdgcn_wmma_f32_16x16x4_f32(false, a, false, b, (short)0, c,
                                               false, false);
}

// ---------------- misc small kernels ----------------

__global__ void zero_acc_kernel(double* acc) {
  if (threadIdx.x < 8) acc[threadIdx.x] = 0.0;
}

__global__ void copy_sigma_kernel(const v4f* __restrict__ src,
                                  v4f* __restrict__ dst, int n4) {
  int stride = gridDim.x * blockDim.x;
  for (int i = blockIdx.x * blockDim.x + threadIdx.x; i < n4; i += stride)
    dst[i] = src[i];
}

__device__ inline void block_reduce_add(float v, double* target) {
  __shared__ float red[256];
  red[threadIdx.x] = v;
  __syncthreads();
  for (int o = 128; o > 0; o >>= 1) {
    if ((int)threadIdx.x < o) red[threadIdx.x] += red[threadIdx.x + o];
    __syncthreads();
  }
  if (threadIdx.x == 0) atomicAdd(target, (double)red[0]);
}

// sum of -0.5 * x^2 over all elements
__global__ void prior_kernel(const v4f* __restrict__ x, int n4, double* acc) {
  float s = 0.f;
  int stride = gridDim.x * blockDim.x;
  for (int i = blockIdx.x * blockDim.x + threadIdx.x; i < n4; i += stride) {
    v4f v = x[i];
    s += v.x * v.x + v.y * v.y + v.z * v.z + v.w * v.w;
  }
  block_reduce_add(-0.5f * s, &acc[0]);
}

// sum of -0.5 * ((mu_pred - y_true[b]) / sigma)^2  (streaming: nontemporal)
__global__ void lik_kernel(const v4f* __restrict__ mu_pred,
                           const float* __restrict__ y_true,
                           const float* __restrict__ sigma_p,
                           int n4, int s_per4, double* acc) {
  float sig = sigma_p[0];
  float inv_s2 = 1.0f / (sig * sig);
  float s = 0.f;
  int stride = gridDim.x * blockDim.x;
  for (int q = blockIdx.x * blockDim.x + threadIdx.x; q < n4; q += stride) {
    int b = q / s_per4;
    float y = y_true[b];
    v4f v = __builtin_nontemporal_load(mu_pred + q);
    float dx = v.x - y, dy = v.y - y, dz = v.z - y, dw = v.w - y;
    s += dx * dx + dy * dy + dz * dz + dw * dw;
  }
  block_reduce_add(-0.5f * inv_s2 * s, &acc[1]);
}

// logdet = sum over factored diagonal tiles of 2*log(diag)
__global__ void logdet_kernel(const float* __restrict__ Dfac, int nblk,
                              double* acc) {
  float s = 0.f;
  int n = nblk * NB;
  for (int t = threadIdx.x; t < n; t += blockDim.x) {
    int kb = t >> 4, j = t & 15;
    s += 2.0f * logf(Dfac[kb * NB * NB + j * NB + j]);
  }
  block_reduce_add(s, &acc[3]);
}

// ------- blocked Cholesky (NB=16, right-looking), diag fused into TRSM -------

// Each block: load raw diag tile, factor it in LDS (column by column, parallel over rows).
// Block 0 publishes the factored tile to Dfac (L's diag blocks stay raw).
// Then one thread per panel row solves X * D^T = A.
__global__ void chol_trsm_kernel(float* L, float* __restrict__ Dfac, int W,
                                 int kb) {
  __shared__ float D[NB][NB];
  int tid = threadIdx.x;
  int base = kb * NB;
  for (int i = tid; i < NB * NB; i += blockDim.x)
    D[i >> 4][i & 15] = L[(size_t)(base + (i >> 4)) * W + base + (i & 15)];
  __syncthreads();
  for (int j = 0; j < NB; ++j) {
    if (tid == j) {
      float s = D[j][j];
      for (int m = 0; m < j; ++m) s -= D[j][m] * D[j][m];
      D[j][j] = sqrtf(s);
    }
    __syncthreads();
    if (tid < NB && tid > j) {
      float t = D[tid][j];
      for (int m = 0; m < j; ++m) t -= D[tid][m] * D[j][m];
      D[tid][j] = t / D[j][j];
    }
    __syncthreads();
  }
  if (blockIdx.x == 0) {
    for (int i = tid; i < NB * NB; i += blockDim.x)
      Dfac[kb * NB * NB + i] = D[i >> 4][i & 15];
  }
  int row = base + NB + blockIdx.x * blockDim.x + tid;
  if (row >= W) return;
  float a[NB];
#pragma unroll
  for (int j = 0; j < NB; ++j) a[j] = L[(size_t)row * W + base + j];
#pragma unroll
  for (int j = 0; j < NB; ++j) {
    float t = a[j];
    for (int m = 0; m < j; ++m) t -= a[m] * D[j][m];
    a[j] = t / D[j][j];
  }
#pragma unroll
  for (int j = 0; j < NB; ++j) L[(size_t)row * W + base + j] = a[j];
}

// trailing SYRK update via f32 WMMA: A[bi,bj] -= P[bi] * P[bj]^T  (1 wave/tile)
__global__ void chol_syrk_kernel(float* L, int W, int kb) {
  int bj = kb + 1 + blockIdx.x;
  int bi = kb + 1 + blockIdx.y;
  if (bj > bi) return;
  int lane = threadIdx.x;
  int r15 = lane & 15;
  int khalf = (lane >> 4) * 2;  // 0 or 2
  int pcol = kb * NB;
  const float* Pi = L + (size_t)(bi * NB + r15) * W + pcol;
  const float* Pj = L + (size_t)(bj * NB + r15) * W + pcol;
  v8f c = {0.f, 0.f, 0.f, 0.f, 0.f, 0.f, 0.f, 0.f};
#pragma unroll
  for (int kc = 0; kc < 4; ++kc) {
    v2f a = *(const v2f*)(Pi + 4 * kc + khalf);
    v2f b = *(const v2f*)(Pj + 4 * kc + khalf);  // B = P[bj]^T: same gather
    c = wmma_f32_k4(a, b, c);
  }
  int colbase = bj * NB;
  int rowbase = bi * NB + ((lane >> 4) ? 8 : 0);
#pragma unroll
  for (int r = 0; r < 8; ++r)
    L[(size_t)(rowbase + r) * W + colbase + r15] -= c[r];
}

// ---- fused forward substitution + maha: Z = L^-1 (noisy - mu)^T, sum z^2 ----
// Forward substitution is column-independent: one wave owns 16 sample columns
// and keeps its whole [W x 16] Z slab in LDS (64 KB of the WGP's 320 KB),
// looping over all 64 block-rows internally. WMMA GEMM pulls B from LDS.
__global__ void fsolve_maha_kernel(const float* __restrict__ noisy,
                                   const float* __restrict__ mu,
                                   const float* __restrict__ L,
                                   const float* __restrict__ Dfac, int W, int S,
                                   int nblk, double* acc) {
  __shared__ float zbuf[MAXW * NB];  // [W][16] column slab
  __shared__ float s_tile[NB * NB];
  int lane = threadIdx.x;
  int r15 = lane & 15;
  int khalf = (lane >> 4) * 2;
  int colbase = blockIdx.x * NB;
  float msum = 0.f;
  for (int kb = 0; kb < nblk; ++kb) {
    const float* Arow = L + (size_t)(kb * NB + r15) * W;
    v8f c = {0.f, 0.f, 0.f, 0.f, 0.f, 0.f, 0.f, 0.f};
    for (int j = 0; j < kb; ++j) {
#pragma unroll
      for (int kc = 0; kc < 4; ++kc) {
        int k = j * NB + 4 * kc + khalf;
        v2f a = *(const v2f*)(Arow + k);
        v2f b;
        b.x = zbuf[(k + 0) * NB + r15];
        b.y = zbuf[(k + 1) * NB + r15];
        c = wmma_f32_k4(a, b, c);
      }
    }
    int mofs = (lane >> 4) ? 8 : 0;
#pragma unroll
    for (int r = 0; r < 8; ++r) s_tile[(r + mofs) * NB + r15] = c[r];
    __syncthreads();
    if (lane < NB) {
      const float* Dt = Dfac + kb * NB * NB;
      int scol = colbase + lane;
      float z[NB];
#pragma unroll
      for (int m = 0; m < NB; ++m) {
        int w = kb * NB + m;
        float rhs = noisy[(size_t)scol * W + w] - mu[w] - s_tile[m * NB + lane];
        for (int mm = 0; mm < m; ++mm) rhs -= Dt[m * NB + mm] * z[mm];
        z[m] = rhs / Dt[m * NB + m];
        zbuf[w * NB + lane] = z[m];
        msum += z[m] * z[m];
      }
    }
    __syncthreads();
  }
  s_tile[lane] = msum;
  __syncthreads();
  if (lane == 0) {
    float t = 0.f;
    for (int i = 0; i < 32; ++i) t += s_tile[i];
    atomicAdd(&acc[2], (double)t);
  }
}

// ---------------- finalize ----------------
__global__ void finalize_kernel(const double* acc, const float* sigma_p,
                                float* out, int S, int W, int B) {
  double prior_sum = acc[0];  // sum -0.5 x^2
  double lik_sum = acc[1];    // sum -0.5 (d/sig)^2
  double maha = acc[2];
  double logdet = acc[3];
  double sig = (double)sigma_p[0];
  double Np = (double)S * (double)W;
  double lp_prior = (prior_sum + Np * (-0.5 * LOG2PI)) / (double)S;  // sigma=1
  double Nl = (double)B * (double)S;
  double lp_lik = (lik_sum + Nl * (-log(sig) - 0.5 * LOG2PI)) / (double)S;
  double lp_var =
      -0.5 * ((double)S * ((double)W * LOG2PI + logdet) + maha) / (double)S;
  double total = (1.0 / 50.0) * (lp_var - lp_prior) - lp_lik;
  out[0] = (float)total;
}

extern "C" void kernel_launch(void* const* d_in, const int* in_sizes, int n_in,
                              void* d_out, int out_size, void* d_ws,
                              size_t ws_size, hipStream_t stream) {
  const float* noisy = (const float*)d_in[0];    // [S, W]
  const float* muw = (const float*)d_in[1];      // [W]
  const float* sigma_m = (const float*)d_in[2];  // [W, W]
  const float* mu_pred = (const float*)d_in[3];  // [B, S]
  const float* sigma_p = (const float*)d_in[4];  // scalar
  const float* y_true = (const float*)d_in[5];   // [B]
  (void)n_in; (void)out_size; (void)ws_size;

  int W = in_sizes[1];
  int S = in_sizes[0] / W;
  int B = in_sizes[5];
  int nblk = W / NB;

  double* acc = (double*)d_ws;
  float* Dfac = (float*)((char*)d_ws + 256);    // [nblk][16][16] factored diags
  float* Lbuf = Dfac + (size_t)nblk * NB * NB;  // [W][W] working copy

  zero_acc_kernel<<<1, 32, 0, stream>>>(acc);
  copy_sigma_kernel<<<512, 256, 0, stream>>>((const v4f*)sigma_m, (v4f*)Lbuf,
                                             W * W / 4);
  prior_kernel<<<256, 256, 0, stream>>>((const v4f*)noisy, S * W / 4, acc);
  lik_kernel<<<2048, 256, 0, stream>>>((const v4f*)mu_pred, y_true, sigma_p,
                                       B * S / 4, S / 4, acc);

  for (int kb = 0; kb < nblk; ++kb) {
    int rows = W - (kb + 1) * NB;
    int g = (rows + 127) / 128;
    if (g < 1) g = 1;  // block 0 still factors + publishes the diag tile
    chol_trsm_kernel<<<g, 128, 0, stream>>>(Lbuf, Dfac, W, kb);
    int nrem = nblk - kb - 1;
    if (nrem > 0) {
      dim3 grid(nrem, nrem);
      chol_syrk_kernel<<<grid, 32, 0, stream>>>(Lbuf, W, kb);
    }
  }

  fsolve_maha_kernel<<<S / NB, 32, 0, stream>>>(noisy, muw, Lbuf, Dfac, W, S,
                                                nblk, acc);
  logdet_kernel<<<1, 256, 0, stream>>>(Dfac, nblk, acc);
  finalize_kernel<<<1, 1, 0, stream>>>(acc, sigma_p, (float*)d_out, S, W, B);
}